// GraphConvPhi_74182675136582
// MI455X (gfx1250) — compile-verified
//
#include <hip/hip_runtime.h>
#include <math.h>

typedef float v2f __attribute__((ext_vector_type(2)));
typedef float v8f __attribute__((ext_vector_type(8)));
typedef int   v4i __attribute__((vector_size(16)));   // matches builtin param type

#define AS1 __attribute__((address_space(1)))
#define AS3 __attribute__((address_space(3)))

#if __has_builtin(__builtin_amdgcn_wmma_f32_16x16x4_f32)
#define HAVE_WMMA_F32 1
#else
#define HAVE_WMMA_F32 0
#endif

#if __has_builtin(__builtin_amdgcn_global_load_async_to_lds_b128)
#define HAVE_ASYNC_LDS 1
#else
#define HAVE_ASYNC_LDS 0
#endif

// ---------------- utility kernels ----------------

__global__ void k_fill0(float* __restrict__ p, long long n) {
    long long i = (long long)blockIdx.x * blockDim.x + threadIdx.x;
    if (i < n) p[i] = 0.0f;
}

__global__ void k_deg(const int* __restrict__ dst, float* __restrict__ deg, int E) {
    int i = blockIdx.x * blockDim.x + threadIdx.x;
    if (i < E) unsafeAtomicAdd(&deg[dst[i]], 1.0f);
}

__global__ void k_invdeg(float* __restrict__ deg, int N) {
    int i = blockIdx.x * blockDim.x + threadIdx.x;
    if (i < N) deg[i] = 1.0f / fmaxf(deg[i], 1.0f);
}

__global__ void k_concat(const float* __restrict__ value, const float* __restrict__ u,
                         float* __restrict__ x, int N) {
    int i = blockIdx.x * blockDim.x + threadIdx.x;
    if (i < N) {
        x[i * 3 + 0] = value[i];
        x[i * 3 + 1] = u[i * 2 + 0];
        x[i * 3 + 2] = u[i * 2 + 1];
    }
}

// ---------------- edge aggregation ----------------

// d = 3 : one thread per edge
__global__ void k_scatter3(const float* __restrict__ x, const int* __restrict__ src,
                           const int* __restrict__ dst, float* __restrict__ agg, int E) {
    int i = blockIdx.x * blockDim.x + threadIdx.x;
    if (i >= E) return;
    int s = src[i], t = dst[i];
    unsafeAtomicAdd(&agg[(size_t)t * 3 + 0], x[(size_t)s * 3 + 0]);
    unsafeAtomicAdd(&agg[(size_t)t * 3 + 1], x[(size_t)s * 3 + 1]);
    unsafeAtomicAdd(&agg[(size_t)t * 3 + 2], x[(size_t)s * 3 + 2]);
}

// d = 1 : one thread per edge (used after pre-multiplying by W10)
__global__ void k_scatter1(const float* __restrict__ z, const int* __restrict__ src,
                           const int* __restrict__ dst, float* __restrict__ tmp, int E) {
    int i = blockIdx.x * blockDim.x + threadIdx.x;
    if (i >= E) return;
    unsafeAtomicAdd(&tmp[dst[i]], z[src[i]]);
}

// d = 64 : one half-wave (16 lanes x float4) per edge; index loads are uniform per half
__global__ void k_scatter64(const float* __restrict__ x, const int* __restrict__ src,
                            const int* __restrict__ dst, float* __restrict__ agg, int E) {
    int h = blockIdx.x * (blockDim.x >> 4) + (threadIdx.x >> 4);  // half-wave id = edge
    int c = threadIdx.x & 15;
    if (h >= E) return;
    int s = src[h], t = dst[h];
    const float4 v = ((const float4*)(x + (size_t)s * 64))[c];
    float* o = agg + (size_t)t * 64 + (size_t)c * 4;
    unsafeAtomicAdd(o + 0, v.x);
    unsafeAtomicAdd(o + 1, v.y);
    unsafeAtomicAdd(o + 2, v.z);
    unsafeAtomicAdd(o + 3, v.w);
}

// d = 128 : one wave (32 lanes x float4) per edge; index loads are uniform per wave
__global__ void k_scatter128(const float* __restrict__ x, const int* __restrict__ src,
                             const int* __restrict__ dst, float* __restrict__ agg, int E) {
    int w = blockIdx.x * (blockDim.x >> 5) + (threadIdx.x >> 5);  // wave id = edge
    int c = threadIdx.x & 31;
    if (w >= E) return;
    int s = src[w], t = dst[w];
    const float4 v = ((const float4*)(x + (size_t)s * 128))[c];
    float* o = agg + (size_t)t * 128 + (size_t)c * 4;
    unsafeAtomicAdd(o + 0, v.x);
    unsafeAtomicAdd(o + 1, v.y);
    unsafeAtomicAdd(o + 2, v.z);
    unsafeAtomicAdd(o + 3, v.w);
}

// ---------------- GEMM: LDS-staged A, WMMA, inv_deg folded into output ----------------

__device__ __forceinline__ float apply_act(float y, int act) {
    if (act == 1) return y > 0.0f ? y : 0.01f * y;   // leaky relu
    if (act == 2) return 1.0f / (1.0f + expf(-y));   // sigmoid
    return y;
}

// out[N,128] = (agg[N,DIN] @ W[DIN,128]) * invdeg[:,None] + b ; block = 16 rows x 128 cols
template <int DIN>
__global__ void k_gemm_wmma(const float* __restrict__ agg, const float* __restrict__ invdeg,
                            const float* __restrict__ W, const float* __restrict__ bias,
                            float* __restrict__ out, int N, int act) {
    constexpr int DOUT   = 128;
    constexpr int STRIDE = DIN + 4;               // pad -> conflict-free ds_load_b64
    __shared__ float As[16 * STRIDE];

    const int tid = threadIdx.x;
    const int m0  = blockIdx.x * 16;

    // ---- stage 16 x DIN A-panel into LDS (shared by all 8 waves) ----
    constexpr int CHUNKS = 16 * DIN / 4;          // float4 chunks
#pragma unroll
    for (int ch = tid; ch < CHUNKS; ch += 256) {
        int r = ch / (DIN / 4);
        int c = ch % (DIN / 4);
        int row = m0 + r;
        if (row >= N) row = N - 1;
        const float* g = agg + (size_t)row * DIN + (size_t)c * 4;
        float* l = &As[r * STRIDE + c * 4];
#if HAVE_ASYNC_LDS
        __builtin_amdgcn_global_load_async_to_lds_b128((AS1 v4i*)g, (AS3 v4i*)l, 0, 0);
#else
        *(float4*)l = *(const float4*)g;
#endif
    }
#if HAVE_ASYNC_LDS
#if __has_builtin(__builtin_amdgcn_s_wait_asynccnt)
    __builtin_amdgcn_s_wait_asynccnt(0);
#else
    asm volatile("s_wait_asynccnt 0" ::: "memory");
#endif
#endif
    __syncthreads();

    const int wave = tid >> 5;
    const int lane = tid & 31;
    const int half = lane >> 4;     // 0 or 1
    const int l16  = lane & 15;
    const int n0   = wave * 16;     // 8 waves cover all 128 output cols

    const float* bptr = W + (size_t)(2 * half) * DOUT + n0 + l16;  // row k0+2*half+r, col n0+l16
    const int aoff = l16 * STRIDE + 2 * half;                      // A row l16, K = k0+2*half+r

#if HAVE_WMMA_F32
    v8f c = {};
#pragma unroll
    for (int k0 = 0; k0 < DIN; k0 += 4) {
        v2f a;
        a.x = As[aoff + k0];
        a.y = As[aoff + k0 + 1];
        v2f b;
        b.x = bptr[(size_t)k0 * DOUT];
        b.y = bptr[(size_t)(k0 + 1) * DOUT];
        c = __builtin_amdgcn_wmma_f32_16x16x4_f32(false, a, false, b, (short)0, c,
                                                  false, false);
    }
    float bv = bias[n0 + l16];
#pragma unroll
    for (int v = 0; v < 8; ++v) {
        int m = m0 + v + 8 * half;
        if (m < N) {
            float y = apply_act(c[v] * invdeg[m] + bv, act);
            out[(size_t)m * DOUT + n0 + l16] = y;
        }
    }
#else
    // scalar fallback (same tiling, no WMMA)
    float acc[8];
#pragma unroll
    for (int v = 0; v < 8; ++v) acc[v] = 0.0f;
    for (int k = 0; k < DIN; ++k) {
        float wv = W[(size_t)k * DOUT + n0 + l16];
#pragma unroll
        for (int v = 0; v < 8; ++v) acc[v] += As[(v + 8 * half) * STRIDE + k] * wv;
    }
    float bv = bias[n0 + l16];
#pragma unroll
    for (int v = 0; v < 8; ++v) {
        int m = m0 + v + 8 * half;
        if (m < N) out[(size_t)m * DOUT + n0 + l16] = apply_act(acc[v] * invdeg[m] + bv, act);
    }
#endif
}

// layer 1: din=3 -> dout=64, no activation
__global__ void k_lin3(const float* __restrict__ agg, const float* __restrict__ invdeg,
                       const float* __restrict__ W, const float* __restrict__ bias,
                       float* __restrict__ out, int N) {
    int i = blockIdx.x * blockDim.x + threadIdx.x;
    if (i >= N * 64) return;
    int n = i >> 6;
    int o = i & 63;
    float s = invdeg[n];
    float a0 = agg[(size_t)n * 3 + 0] * s;
    float a1 = agg[(size_t)n * 3 + 1] * s;
    float a2 = agg[(size_t)n * 3 + 2] * s;
    out[i] = fmaf(a0, W[o], fmaf(a1, W[64 + o], fmaf(a2, W[128 + o], bias[o])));
}

// layer 10 pre-multiply: z[n] = h9[n] . W10  (no scaling / bias here)
__global__ void k_dot128(const float* __restrict__ x, const float* __restrict__ W,
                         float* __restrict__ z, int N) {
    int n = blockIdx.x * blockDim.x + threadIdx.x;
    if (n >= N) return;
    const float4* row = (const float4*)(x + (size_t)n * 128);
    const float4* w   = (const float4*)W;
    float acc = 0.0f;
#pragma unroll
    for (int j = 0; j < 32; ++j) {
        float4 v = row[j];
        float4 ww = w[j];
        acc = fmaf(v.x, ww.x, acc);
        acc = fmaf(v.y, ww.y, acc);
        acc = fmaf(v.z, ww.z, acc);
        acc = fmaf(v.w, ww.w, acc);
    }
    z[n] = acc;
}

__global__ void k_finalize(const float* __restrict__ tmp, const float* __restrict__ invdeg,
                           const float* __restrict__ bias, float* __restrict__ out, int N) {
    int n = blockIdx.x * blockDim.x + threadIdx.x;
    if (n < N) out[n] = tmp[n] * invdeg[n] + bias[0];
}

// ---------------- host orchestration ----------------

static inline int cdiv_ll(long long a, long long b) { return (int)((a + b - 1) / b); }

extern "C" void kernel_launch(void* const* d_in, const int* in_sizes, int n_in,
                              void* d_out, int out_size, void* d_ws, size_t ws_size,
                              hipStream_t stream) {
    const int N = in_sizes[0];   // value: [N,1]
    const int E = in_sizes[2];   // src: [E]
    const float* value = (const float*)d_in[0];
    const float* u     = (const float*)d_in[1];
    const int*   src   = (const int*)d_in[2];
    const int*   dst   = (const int*)d_in[3];
    const float* W[10];
    const float* B[10];
    for (int i = 0; i < 10; ++i) {
        W[i] = (const float*)d_in[4 + 2 * i];
        B[i] = (const float*)d_in[5 + 2 * i];
    }

    float* bufX   = (float*)d_ws;                 // node features [N,<=128]
    float* bufA   = bufX + (size_t)N * 128;       // aggregation scratch [N,<=128]
    float* invdeg = bufA + (size_t)N * 128;       // [N]

    const int T = 256;

    // in-degree -> inv_deg (clamped at 1)
    k_fill0<<<cdiv_ll(N, T), T, 0, stream>>>(invdeg, N);
    k_deg<<<cdiv_ll(E, T), T, 0, stream>>>(dst, invdeg, E);
    k_invdeg<<<cdiv_ll(N, T), T, 0, stream>>>(invdeg, N);

    // x0 = concat(value, u)  [N,3]
    k_concat<<<cdiv_ll(N, T), T, 0, stream>>>(value, u, bufX, N);

    // layer 1: 3 -> 64
    k_fill0<<<cdiv_ll((long long)N * 3, T), T, 0, stream>>>(bufA, (long long)N * 3);
    k_scatter3<<<cdiv_ll(E, T), T, 0, stream>>>(bufX, src, dst, bufA, E);
    k_lin3<<<cdiv_ll((long long)N * 64, T), T, 0, stream>>>(bufA, invdeg, W[0], B[0], bufX, N);

    const int mblocks = (N + 15) / 16;

    // layer 2: 64 -> 128 (leaky relu)
    k_fill0<<<cdiv_ll((long long)N * 64, T), T, 0, stream>>>(bufA, (long long)N * 64);
    k_scatter64<<<cdiv_ll(E, 16), T, 0, stream>>>(bufX, src, dst, bufA, E);
    k_gemm_wmma<64><<<mblocks, T, 0, stream>>>(bufA, invdeg, W[1], B[1], bufX, N, 1);

    // layers 3..9: 128 -> 128
    const int actc[7] = {1, 1, 0, 0, 0, 0, 2};
    for (int i = 2; i <= 8; ++i) {
        k_fill0<<<cdiv_ll((long long)N * 128, T), T, 0, stream>>>(bufA, (long long)N * 128);
        k_scatter128<<<cdiv_ll(E, 8), T, 0, stream>>>(bufX, src, dst, bufA, E);
        k_gemm_wmma<128><<<mblocks, T, 0, stream>>>(bufA, invdeg, W[i], B[i], bufX, N, actc[i - 2]);
    }

    // layer 10: 128 -> 1, via linearity: z = h9 @ W10, scatter scalars, finalize
    float* z   = bufA;          // [N]
    float* tmp = bufA + N;      // [N]
    k_dot128<<<cdiv_ll(N, T), T, 0, stream>>>(bufX, W[9], z, N);
    k_fill0<<<cdiv_ll(N, T), T, 0, stream>>>(tmp, N);
    k_scatter1<<<cdiv_ll(E, T), T, 0, stream>>>(z, src, dst, tmp, E);
    k_finalize<<<cdiv_ll(N, T), T, 0, stream>>>(tmp, invdeg, B[9], (float*)d_out, N);
}